// FusedTensorProductOp4_52879637348701
// MI455X (gfx1250) — compile-verified
//
#include <hip/hip_runtime.h>
#include <cstdint>
#include <cmath>

// ---------------------------------------------------------------------------
// Segmented tensor product 'uvw,iu,jv,kw' on MI455X (gfx1250, wave32, WMMA).
//
// out[z,k,w] = sum_{i,j,u,v} cg[i,j,k] * W_p[u,v,w] * x1[z,i,u] * x2[z,j,v]
//
// Factorized per path as a GEMM with K=4096 over (u,v):
//   X2c_i[z,k,v] = sum_j cg[i,j,k] * x2[z,j,v]          (tiny, VALU)
//   A[(z),(u,v)] = sum_i x1[z,i,u] * X2c_i[z,k,v]       (built on the fly)
//   out[(z,k),w] += A . W_p[(u,v),w]                    (V_WMMA_F32_16X16X4_F32)
//
// v1 -> v2: t-loop unrolled only 4x (was 16x). The full unroll pushed the
// wave past 256 VGPRs (s_set_vgpr_msb storm in the disasm, ~35% of all
// instructions were SALU overhead). Partial unroll keeps ~55 live VGPR
// temporaries + 40 accumulators, comfortably under 256, restoring occupancy
// while still keeping 4 global B-loads in flight for latency hiding.
// ---------------------------------------------------------------------------

#define MULTC 64
#define W3C   (MULTC * MULTC * MULTC)
#define FEAT  576
#define NPATH 15
#define ZB    16   // z rows per block tile (WMMA M)

typedef float v2f __attribute__((ext_vector_type(2)));
typedef float v8f __attribute__((ext_vector_type(8)));

// -------------------------- CG generation (device) -------------------------
// Exact reproduction of np.random.RandomState(0).standard_normal sequence:
// MT19937 with Knuth scalar seeding (mt[0]=0) + legacy polar-method Gaussian.
__global__ void cg_init_kernel(float* __restrict__ cg) {
  if (threadIdx.x != 0 || blockIdx.x != 0) return;

  unsigned mt[624];
  int mti;
  mt[0] = 0u;
  for (int i = 1; i < 624; ++i)
    mt[i] = 1812433253u * (mt[i - 1] ^ (mt[i - 1] >> 30)) + (unsigned)i;
  mti = 624;

  bool has_g = false;
  double gcache = 0.0;

  const int pl1[NPATH] = {0,0,0,1,1,1,1,1,1,2,2,2,2,2,2};
  const int pl2[NPATH] = {0,1,2,0,1,1,1,2,2,0,1,1,2,2,2};
  const int pl3[NPATH] = {0,1,2,1,0,1,2,1,2,2,1,2,0,1,2};

  int off = 0;
  for (int p = 0; p < NPATH; ++p) {
    const int I = 2 * pl1[p] + 1, J = 2 * pl2[p] + 1, K = 2 * pl3[p] + 1;
    const double scale = 1.0 / sqrt((double)(I * J * MULTC * MULTC));
    const int n = I * J * K;
    for (int e = 0; e < n; ++e) {
      double g;
      if (has_g) { g = gcache; has_g = false; }
      else {
        double a1, a2, r2;
        do {
          double u[2];
          for (int s = 0; s < 2; ++s) {
            unsigned hi = 0, lo = 0;
            for (int d = 0; d < 2; ++d) {
              if (mti >= 624) {
                for (int i = 0; i < 624; ++i) {
                  unsigned y = (mt[i] & 0x80000000u) | (mt[(i + 1) % 624] & 0x7fffffffu);
                  mt[i] = mt[(i + 397) % 624] ^ (y >> 1) ^ ((y & 1u) ? 0x9908b0dfu : 0u);
                }
                mti = 0;
              }
              unsigned y = mt[mti++];
              y ^= y >> 11; y ^= (y << 7) & 0x9d2c5680u;
              y ^= (y << 15) & 0xefc60000u; y ^= y >> 18;
              if (d == 0) hi = y; else lo = y;
            }
            u[s] = ((double)(hi >> 5) * 67108864.0 + (double)(lo >> 6)) / 9007199254740992.0;
          }
          a1 = 2.0 * u[0] - 1.0;
          a2 = 2.0 * u[1] - 1.0;
          r2 = a1 * a1 + a2 * a2;
        } while (r2 >= 1.0 || r2 == 0.0);
        double f = sqrt(-2.0 * log(r2) / r2);
        gcache = f * a1; has_g = true;
        g = f * a2;
      }
      cg[off + e] = (float)(g * scale);
    }
    off += n;
  }
}

// ------------------------------ output zeroing ------------------------------
__global__ void zero_kernel(float* __restrict__ out, int n) {
  int i = blockIdx.x * blockDim.x + threadIdx.x;
  if (i < n) out[i] = 0.0f;
}

// ------------------------------ main TP kernel ------------------------------
template <int L1, int L2, int L3, int P, int CGO>
__device__ __forceinline__ void tp_body(const float* __restrict__ x0,
                                        const float* __restrict__ x1,
                                        const float* __restrict__ x2,
                                        const float* __restrict__ cg,
                                        float* __restrict__ out,
                                        float* smem) {
  constexpr int I = 2 * L1 + 1, J = 2 * L2 + 1, K = 2 * L3 + 1;
  constexpr int S1 = (L1 == 0) ? 0 : (L1 == 1) ? 64 : 256;
  constexpr int S2 = (L2 == 0) ? 0 : (L2 == 1) ? 64 : 256;
  constexpr int S3 = (L3 == 0) ? 0 : (L3 == 1) ? 64 : 256;

  float* x1t = smem;                 // [I*64][ZB]   x1 segment, [feat][z]
  float* x2t = x1t + I * 64 * ZB;    // [J*64][ZB]
  float* xct = x2t + J * 64 * ZB;    // [(i*K+k)*64+v][ZB]  X2c

  const int tid = (int)threadIdx.x;
  const int z0  = (int)blockIdx.x * ZB;

  // Stage x1/x2 z-tile segments transposed into LDS (coalesced global reads).
  for (int idx = tid; idx < I * 64 * ZB; idx += 128) {
    int z = idx / (I * 64), c = idx % (I * 64);
    x1t[c * ZB + z] = x1[(z0 + z) * FEAT + S1 + c];
  }
  for (int idx = tid; idx < J * 64 * ZB; idx += 128) {
    int z = idx / (J * 64), c = idx % (J * 64);
    x2t[c * ZB + z] = x2[(z0 + z) * FEAT + S2 + c];
  }
  __syncthreads();

  // X2c_i[z,k,v] = sum_j cg[i,j,k] * x2[z,j,v]
  for (int idx = tid; idx < I * K * 64 * ZB; idx += 128) {
    const int z  = idx & (ZB - 1);
    const int v  = (idx >> 4) & 63;
    const int ik = idx >> 10;           // i*K + k
    const int i  = ik / K, k = ik % K;
    float s = 0.0f;
#pragma unroll
    for (int j = 0; j < J; ++j)
      s += cg[CGO + (i * J + j) * K + k] * x2t[(j * 64 + v) * ZB + z];
    xct[idx] = s;
  }
  __syncthreads();

  const int lane = tid & 31;
  const int wv   = tid >> 5;                  // wave = N-tile of w
  const int zl   = lane & 15;                 // z row owned by lane (A/C M-dim)
  const int c0   = (lane < 16) ? 0 : 2;       // K sub-index pair for A/B VGPRs
  const int wcol = wv * 16 + (lane & 15);     // output w column (B/C N-dim)
  const float* Wp = x0 + (long)P * W3C;

  v8f acc[K];
#pragma unroll
  for (int k = 0; k < K; ++k) acc[k] = v8f{0, 0, 0, 0, 0, 0, 0, 0};

  for (int v = 0; v < 64; ++v) {
    float xc[I][K];
#pragma unroll
    for (int i = 0; i < I; ++i)
#pragma unroll
      for (int k = 0; k < K; ++k)
        xc[i][k] = xct[((i * K + k) * 64 + v) * ZB + zl];

    // Partial unroll: 4 K-steps live at once -> ~4 B-loads in flight but
    // per-wave VGPR footprint stays well under 256 (no VGPR-MSB thrash).
#pragma unroll 4
    for (int t = 0; t < 16; ++t) {          // K-steps over u (4 at a time)
      const int u0 = 4 * t;
      float xa[I], xb[I];
#pragma unroll
      for (int i = 0; i < I; ++i) {
        xa[i] = x1t[(i * 64 + u0 + c0) * ZB + zl];
        xb[i] = x1t[(i * 64 + u0 + c0 + 1) * ZB + zl];
      }
      // B operand: W_p[(u,v), w] for this wave's 16 w-columns (L2-resident)
      v2f B;
      B.x = Wp[(u0 + c0) * 4096 + v * 64 + wcol];
      B.y = Wp[(u0 + c0 + 1) * 4096 + v * 64 + wcol];
#pragma unroll
      for (int k = 0; k < K; ++k) {
        float a0 = 0.0f, a1 = 0.0f;
#pragma unroll
        for (int i = 0; i < I; ++i) {
          a0 += xa[i] * xc[i][k];
          a1 += xb[i] * xc[i][k];
        }
        v2f A; A.x = a0; A.y = a1;
        acc[k] = __builtin_amdgcn_wmma_f32_16x16x4_f32(
            false, A, false, B, (short)0, acc[k], false, false);
      }
    }
  }

  // C/D layout: VGPR r holds M=r (lanes 0-15) / M=r+8 (lanes 16-31), N=lane%16.
  const int m0 = (lane < 16) ? 0 : 8;
#pragma unroll
  for (int k = 0; k < K; ++k)
#pragma unroll
    for (int r = 0; r < 8; ++r)
      unsafeAtomicAdd(&out[(z0 + m0 + r) * FEAT + S3 + k * 64 + wcol], acc[k][r]);
}

__global__ __launch_bounds__(128) void tp_kernel(const float* __restrict__ x0,
                                                 const float* __restrict__ x1,
                                                 const float* __restrict__ x2,
                                                 const float* __restrict__ cg,
                                                 float* __restrict__ out) {
  extern __shared__ float smem[];
  switch (blockIdx.y) {
    case 0:  tp_body<0,0,0, 0,  0 >(x0, x1, x2, cg, out, smem); break;
    case 1:  tp_body<0,1,1, 1,  1 >(x0, x1, x2, cg, out, smem); break;
    case 2:  tp_body<0,2,2, 2,  10>(x0, x1, x2, cg, out, smem); break;
    case 3:  tp_body<1,0,1, 3,  35>(x0, x1, x2, cg, out, smem); break;
    case 4:  tp_body<1,1,0, 4,  44>(x0, x1, x2, cg, out, smem); break;
    case 5:  tp_body<1,1,1, 5,  53>(x0, x1, x2, cg, out, smem); break;
    case 6:  tp_body<1,1,2, 6,  80>(x0, x1, x2, cg, out, smem); break;
    case 7:  tp_body<1,2,1, 7, 125>(x0, x1, x2, cg, out, smem); break;
    case 8:  tp_body<1,2,2, 8, 170>(x0, x1, x2, cg, out, smem); break;
    case 9:  tp_body<2,0,2, 9, 245>(x0, x1, x2, cg, out, smem); break;
    case 10: tp_body<2,1,1,10, 270>(x0, x1, x2, cg, out, smem); break;
    case 11: tp_body<2,1,2,11, 315>(x0, x1, x2, cg, out, smem); break;
    case 12: tp_body<2,2,0,12, 390>(x0, x1, x2, cg, out, smem); break;
    case 13: tp_body<2,2,1,13, 415>(x0, x1, x2, cg, out, smem); break;
    case 14: tp_body<2,2,2,14, 490>(x0, x1, x2, cg, out, smem); break;
  }
}

// ------------------------------- launcher -----------------------------------
extern "C" void kernel_launch(void* const* d_in, const int* in_sizes, int n_in,
                              void* d_out, int out_size, void* d_ws, size_t ws_size,
                              hipStream_t stream) {
  const float* x0 = (const float*)d_in[0];
  const float* x1 = (const float*)d_in[1];
  const float* x2 = (const float*)d_in[2];
  float* out = (float*)d_out;
  float* cg  = (float*)d_ws;   // 615 floats of CG coefficients

  // 1) regenerate CG constants (deterministic, capture-safe)
  cg_init_kernel<<<1, 1, 0, stream>>>(cg);

  // 2) zero the accumulation buffer
  zero_kernel<<<(out_size + 255) / 256, 256, 0, stream>>>(out, out_size);

  // 3) main fused tensor product
  const int Z = in_sizes[1] / FEAT;          // 2048
  dim3 grid(Z / ZB, NPATH);                  // (128, 15)
  // LDS sized for the largest path (l1=l2=l3=2): x1t + x2t + X2c
  size_t smem = (size_t)(5 * 64 * ZB + 5 * 64 * ZB + 5 * 5 * 64 * ZB) * sizeof(float);
  tp_kernel<<<grid, 128, smem, stream>>>(x0, x1, x2, cg, out);
}